// TransformerXLAttention_26903675142803
// MI455X (gfx1250) — compile-verified
//
#include <hip/hip_runtime.h>
#include <hip/hip_bf16.h>
#include <math.h>

// ---------------------------------------------------------------------------
// CDNA5 (gfx1250) Transformer-XL attention, fp32, via V_WMMA_F32_16X16X4_F32.
// wave32 only. Each wave computes a 32x64 (or 16x64) C tile; 8 v_wmma per
// 4-deep K step. All GEMM operands fit in the 192MB L2, score matrix is
// processed in 4 slab-batches of 8 heads (64MB live) to bound workspace.
// ---------------------------------------------------------------------------

typedef float v2f __attribute__((ext_vector_type(2)));
typedef float v8f __attribute__((ext_vector_type(8)));

__device__ __forceinline__ v8f v8f_zero() {
    v8f z = {0.f, 0.f, 0.f, 0.f, 0.f, 0.f, 0.f, 0.f};
    return z;
}

// D = A(16x4) * B(4x16) + C, fp32 matrix pipe.
// args: (neg_a, A, neg_b, B, c_mod, C, reuse_a, reuse_b)
__device__ __forceinline__ v8f wmma4(v2f a, v2f b, v8f c) {
    return __builtin_amdgcn_wmma_f32_16x16x4_f32(false, a, false, b,
                                                 (short)0, c, false, false);
}

// ---------------------------------------------------------------------------
// Kernel 1: flipped sinusoidal relative position embedding, (2048 x 1024).
// rel[k][d] = pe[2047-k][d];  pe[:, :512]=sin(pos*invf), pe[:,512:]=cos.
// ---------------------------------------------------------------------------
__global__ void relgen_kernel(float* __restrict__ rel) {
    const int row = blockIdx.x;          // k in 0..2047
    const int t = threadIdx.x;           // 256 threads
    const float pos = (float)(2047 - row);
    const float negl = -9.210340371976184f / 512.0f;  // -ln(10000)/half
#pragma unroll
    for (int i = 0; i < 4; ++i) {
        int c = t + i * 256;             // 0..1023
        int cc = c & 511;
        float inv = expf((float)cc * negl);
        float ang = pos * inv;
        rel[(size_t)row * 1024 + c] = (c < 512) ? sinf(ang) : cosf(ang);
    }
}

// ---------------------------------------------------------------------------
// Kernel 2: generic NT GEMM  C[M,N] = A[M,Kd] * B[N,Kd]^T  (all row-major).
// Block = 256 threads = 8 waves. Block tile 64(M) x 256(N); wave tile 32x64.
// Requires M%64==0, N%256==0, Kd%4==0.
// ---------------------------------------------------------------------------
__global__ void gemm_nt_kernel(const float* __restrict__ A,
                               const float* __restrict__ B,
                               float* __restrict__ C,
                               int M, int N, int Kd) {
    const int lane = threadIdx.x & 31;
    const int w    = threadIdx.x >> 5;
    const int l16  = lane & 15;
    const int hlf  = lane >> 4;                 // 0 or 1
    const int m0 = blockIdx.y * 64 + (w & 1) * 32;
    const int n0 = blockIdx.x * 256 + (w >> 1) * 64;
    (void)M;

    v8f c[2][4];
#pragma unroll
    for (int i = 0; i < 2; ++i)
#pragma unroll
        for (int j = 0; j < 4; ++j) c[i][j] = v8f_zero();

    for (int kb = 0; kb < Kd; kb += 4) {
        const int kk = kb + 2 * hlf;            // A/B frag K pair per half-wave
        v2f a[2], b[4];
#pragma unroll
        for (int i = 0; i < 2; ++i)
            a[i] = *(const v2f*)(A + (size_t)(m0 + i * 16 + l16) * Kd + kk);
#pragma unroll
        for (int j = 0; j < 4; ++j)
            b[j] = *(const v2f*)(B + (size_t)(n0 + j * 16 + l16) * Kd + kk);
#pragma unroll
        for (int i = 0; i < 2; ++i)
#pragma unroll
            for (int j = 0; j < 4; ++j) c[i][j] = wmma4(a[i], b[j], c[i][j]);
    }

#pragma unroll
    for (int i = 0; i < 2; ++i)
#pragma unroll
        for (int j = 0; j < 4; ++j)
#pragma unroll
            for (int r = 0; r < 8; ++r) {
                int m = m0 + i * 16 + r + 8 * hlf;
                int n = n0 + j * 16 + l16;
                C[(size_t)m * N + n] = c[i][j][r];
            }
}

// ---------------------------------------------------------------------------
// Kernel 3: attention scores for 8 heads (slab batch).
//   S[q,k] = ((Q+u).K^T + (Q+v).R^T) / 8, with causal mask k > 1024+q -> -inf.
// Q: (B,1024,16,64); K: (B,2048,16,64); R: (2048,16,64); u,v: (16,64).
// Grid: (2048/256, 1024/64, 8); block tile 64(q) x 256(k); wave tile 32x64.
// ---------------------------------------------------------------------------
__global__ void score_kernel(const float* __restrict__ Q,
                             const float* __restrict__ Km,
                             const float* __restrict__ R,
                             const float* __restrict__ u,
                             const float* __restrict__ v,
                             float* __restrict__ S, int bh0) {
    const int lane = threadIdx.x & 31;
    const int w    = threadIdx.x >> 5;
    const int l16  = lane & 15;
    const int hlf  = lane >> 4;
    const int s  = blockIdx.z;
    const int bh = bh0 + s;
    const int b  = bh >> 4;
    const int h  = bh & 15;
    const int m0 = blockIdx.y * 64 + (w & 1) * 32;    // q
    const int n0 = blockIdx.x * 256 + (w >> 1) * 64;  // k

    const float* Qb = Q + (size_t)b * 1024 * 1024 + h * 64;  // row stride 1024
    const float* Kb = Km + (size_t)b * 2048 * 1024 + h * 64; // row stride 1024
    const float* Rb = R + h * 64;                            // row stride 1024

    v8f c[2][4];
#pragma unroll
    for (int i = 0; i < 2; ++i)
#pragma unroll
        for (int j = 0; j < 4; ++j) c[i][j] = v8f_zero();

    for (int kb = 0; kb < 64; kb += 4) {
        const int kk = kb + 2 * hlf;
        v2f ub = *(const v2f*)(u + h * 64 + kk);
        v2f vb = *(const v2f*)(v + h * 64 + kk);
        v2f qa[2], qv[2], bk[4], br[4];
#pragma unroll
        for (int i = 0; i < 2; ++i) {
            v2f qq = *(const v2f*)(Qb + (size_t)(m0 + i * 16 + l16) * 1024 + kk);
            qa[i] = qq + ub;
            qv[i] = qq + vb;
        }
#pragma unroll
        for (int j = 0; j < 4; ++j) {
            bk[j] = *(const v2f*)(Kb + (size_t)(n0 + j * 16 + l16) * 1024 + kk);
            br[j] = *(const v2f*)(Rb + (size_t)(n0 + j * 16 + l16) * 1024 + kk);
        }
#pragma unroll
        for (int i = 0; i < 2; ++i)
#pragma unroll
            for (int j = 0; j < 4; ++j) c[i][j] = wmma4(qa[i], bk[j], c[i][j]);
#pragma unroll
        for (int i = 0; i < 2; ++i)
#pragma unroll
            for (int j = 0; j < 4; ++j) c[i][j] = wmma4(qv[i], br[j], c[i][j]);
    }

    const float scale = 0.125f;  // 1/sqrt(64)
    float* Ss = S + (size_t)s * 1024 * 2048;
#pragma unroll
    for (int i = 0; i < 2; ++i)
#pragma unroll
        for (int j = 0; j < 4; ++j)
#pragma unroll
            for (int r = 0; r < 8; ++r) {
                int m = m0 + i * 16 + r + 8 * hlf;  // q
                int n = n0 + j * 16 + l16;          // k
                float val = c[i][j][r] * scale;
                if (n > 1024 + m) val = -INFINITY;  // L_kv - L_q + q
                Ss[(size_t)m * 2048 + n] = val;
            }
}

// ---------------------------------------------------------------------------
// Kernel 4: in-place row softmax over 2048 elements. Grid (1024, 8), 256 thr.
// ---------------------------------------------------------------------------
__global__ void softmax_kernel(float* __restrict__ S) {
    __shared__ float red[256];
    const int s = blockIdx.y, q = blockIdx.x, t = threadIdx.x;
    float* row = S + ((size_t)s * 1024 + q) * 2048;
    float x[8];
    float m = -INFINITY;
#pragma unroll
    for (int i = 0; i < 8; ++i) {
        x[i] = row[t + i * 256];
        m = fmaxf(m, x[i]);
    }
    red[t] = m;
    __syncthreads();
    for (int off = 128; off > 0; off >>= 1) {
        if (t < off) red[t] = fmaxf(red[t], red[t + off]);
        __syncthreads();
    }
    m = red[0];
    __syncthreads();
    float sum = 0.f;
#pragma unroll
    for (int i = 0; i < 8; ++i) {
        x[i] = __expf(x[i] - m);
        sum += x[i];
    }
    red[t] = sum;
    __syncthreads();
    for (int off = 128; off > 0; off >>= 1) {
        if (t < off) red[t] += red[t + off];
        __syncthreads();
    }
    float inv = 1.0f / red[0];
#pragma unroll
    for (int i = 0; i < 8; ++i) row[t + i * 256] = x[i] * inv;
}

// ---------------------------------------------------------------------------
// Kernel 5: O[q,d] = sum_k P[q,k] * V[b,k,h,d] for 8 heads.
// Grid (1024/128, 8), 256 thr = 8 waves; each wave a 16(q) x 64(d) tile.
// Writes O in (B, L_q, H, hd) layout so the out-projection is a plain NT GEMM.
// ---------------------------------------------------------------------------
__global__ void pv_kernel(const float* __restrict__ P,
                          const float* __restrict__ V,
                          float* __restrict__ O, int bh0) {
    const int lane = threadIdx.x & 31;
    const int w    = threadIdx.x >> 5;
    const int l16  = lane & 15;
    const int hlf  = lane >> 4;
    const int s  = blockIdx.y;
    const int bh = bh0 + s;
    const int b  = bh >> 4;
    const int h  = bh & 15;
    const int m0 = blockIdx.x * 128 + w * 16;  // q tile

    const float* Ps = P + (size_t)s * 1024 * 2048;
    const float* Vb = V + (size_t)b * 2048 * 1024 + h * 64;  // row stride 1024

    v8f c[4];
#pragma unroll
    for (int j = 0; j < 4; ++j) c[j] = v8f_zero();

    for (int kb = 0; kb < 2048; kb += 4) {
        const int kk = kb + 2 * hlf;
        v2f a = *(const v2f*)(Ps + (size_t)(m0 + l16) * 2048 + kk);
        v2f bf[4];
#pragma unroll
        for (int j = 0; j < 4; ++j) {
            int n = j * 16 + l16;
            bf[j].x = Vb[(size_t)kk * 1024 + n];
            bf[j].y = Vb[(size_t)(kk + 1) * 1024 + n];
        }
#pragma unroll
        for (int j = 0; j < 4; ++j) c[j] = wmma4(a, bf[j], c[j]);
    }

#pragma unroll
    for (int j = 0; j < 4; ++j)
#pragma unroll
        for (int r = 0; r < 8; ++r) {
            int m = m0 + r + 8 * hlf;
            int n = j * 16 + l16;
            O[((size_t)b * 1024 + m) * 1024 + h * 64 + n] = c[j][r];
        }
}

// ---------------------------------------------------------------------------
// Host-side launcher.
// Workspace layout (floats): Q 2M | K 4M | V 4M | rel 2M | R 2M | O 2M | S 16M
// = 32M floats = 128 MB total.
// ---------------------------------------------------------------------------
extern "C" void kernel_launch(void* const* d_in, const int* in_sizes, int n_in,
                              void* d_out, int out_size, void* d_ws, size_t ws_size,
                              hipStream_t stream) {
    const float* x_q  = (const float*)d_in[0];
    const float* x_kv = (const float*)d_in[1];
    const float* Wq   = (const float*)d_in[2];
    const float* Wk   = (const float*)d_in[3];
    const float* Wv   = (const float*)d_in[4];
    const float* Wr   = (const float*)d_in[5];
    const float* Wo   = (const float*)d_in[6];
    const float* ub   = (const float*)d_in[7];
    const float* vb   = (const float*)d_in[8];
    float* out = (float*)d_out;
    (void)in_sizes; (void)n_in; (void)out_size; (void)ws_size;

    float* ws   = (float*)d_ws;
    float* Qb   = ws;                    // 2,097,152
    float* Kb   = Qb + 2097152;          // 4,194,304
    float* Vb   = Kb + 4194304;          // 4,194,304
    float* Relb = Vb + 4194304;          // 2,097,152
    float* Rb   = Relb + 2097152;        // 2,097,152
    float* Ob   = Rb + 2097152;          // 2,097,152
    float* Sb   = Ob + 2097152;          // 16,777,216

    // Relative position embeddings (flipped).
    relgen_kernel<<<2048, 256, 0, stream>>>(Relb);

    // Projections: X @ W^T (NT GEMM).
    gemm_nt_kernel<<<dim3(4, 32), 256, 0, stream>>>(x_q,  Wq, Qb, 2048, 1024, 1024);
    gemm_nt_kernel<<<dim3(4, 64), 256, 0, stream>>>(x_kv, Wk, Kb, 4096, 1024, 1024);
    gemm_nt_kernel<<<dim3(4, 64), 256, 0, stream>>>(x_kv, Wv, Vb, 4096, 1024, 1024);
    gemm_nt_kernel<<<dim3(4, 32), 256, 0, stream>>>(Relb, Wr, Rb, 2048, 1024, 1024);

    // Attention in 4 slab-batches of 8 (b,h) heads (bounds S to 64 MB live).
    for (int bh0 = 0; bh0 < 32; bh0 += 8) {
        score_kernel<<<dim3(8, 16, 8), 256, 0, stream>>>(Qb, Kb, Rb, ub, vb, Sb, bh0);
        softmax_kernel<<<dim3(1024, 8), 256, 0, stream>>>(Sb);
        pv_kernel<<<dim3(8, 8), 256, 0, stream>>>(Sb, Vb, Ob, bh0);
    }

    // Output projection: O @ Wo^T.
    gemm_nt_kernel<<<dim3(4, 32), 256, 0, stream>>>(Ob, Wo, out, 2048, 1024, 1024);
}